// PlaceFields_78932908966357
// MI455X (gfx1250) — compile-verified
//
#include <hip/hip_runtime.h>
#include <hip/hip_bf16.h>

typedef __attribute__((ext_vector_type(2))) float v2f;
typedef __attribute__((ext_vector_type(8))) float v8f;

#define NPIX 65536   // H*W = 256*256
#define NG   512     // number of gaussians
#define GSTR 528     // LDS stride for feature planes (not multiple of 64 -> no bank conflicts)

// Each wave computes 16-pixel x 16-gaussian tiles of q = F_p . G_n via two
// chained V_WMMA_F32_16X16X4_F32 (K padded 6->8), then pred/err/accumulate.
__global__ __launch_bounds__(256) void PlaceFields_78932908966357_kernel(
    const float* __restrict__ coords,        // [NPIX,2]
    const float* __restrict__ realv,         // [NG,NPIX]
    const float* __restrict__ means,         // [NG,2]
    const float* __restrict__ cov_inv_diag,  // [NG,2]
    const float* __restrict__ cov_inv_off,   // [NG,1]
    const float* __restrict__ scales,        // [NG,1]
    float* __restrict__ ws)                  // [gridDim.x] block partials
{
    __shared__ float Gf[8 * GSTR];   // feature planes k=0..7, SoA
    __shared__ float Sc[NG];
    __shared__ float red[8];

    const int t = threadIdx.x;

    // ---- Precompute gaussian features into LDS (once per block) ----
    for (int n = t; n < NG; n += 256) {
        float a  = cov_inv_diag[2 * n];
        float c  = cov_inv_diag[2 * n + 1];
        float b  = cov_inv_off[n];
        float mx = means[2 * n];
        float my = means[2 * n + 1];
        Gf[0 * GSTR + n] = a;
        Gf[1 * GSTR + n] = 2.0f * b;
        Gf[2 * GSTR + n] = c;
        Gf[3 * GSTR + n] = -2.0f * (a * mx + b * my);
        Gf[4 * GSTR + n] = -2.0f * (b * mx + c * my);
        Gf[5 * GSTR + n] = a * mx * mx + 2.0f * b * mx * my + c * my * my;
        Gf[6 * GSTR + n] = 0.0f;
        Gf[7 * GSTR + n] = 0.0f;
        Sc[n] = scales[n];
    }
    __syncthreads();

    const int lane = t & 31;
    const int wave = t >> 5;
    const int n16  = lane & 15;       // N index within tile (pixel)
    const int half = lane >> 4;       // 0: K=0,1  1: K=2,3 (A/B operand halves)
    const int kb   = half * 2;

    const int p0 = (blockIdx.x * 8 + wave) * 16;   // pixel tile base
    const int p  = p0 + n16;

    // ---- Pixel features (B operand, 4x16 per WMMA, K padded with zeros) ----
    const float x = coords[2 * p];
    const float y = coords[2 * p + 1];
    // F = [x*x, x*y, y*y, x, y, 1, 0, 0]
    v2f b0, b1;
    b0.x = half ? (y * y) : (x * x);   // K = kb
    b0.y = half ? x       : (x * y);   // K = kb+1
    b1.x = half ? 0.0f    : y;         // K = 4+kb
    b1.y = half ? 0.0f    : 1.0f;      // K = 5+kb

    float acc = 0.0f;

    for (int g0 = 0; g0 < NG; g0 += 16) {
        // ---- A operand: 16 gaussians x K (lane m = n16, K per lane-half) ----
        const int gm = g0 + n16;
        v2f a0, a1;
        a0.x = Gf[(kb)     * GSTR + gm];
        a0.y = Gf[(kb + 1) * GSTR + gm];
        a1.x = Gf[(kb + 4) * GSTR + gm];
        a1.y = Gf[(kb + 5) * GSTR + gm];

        v8f q = {};
        q = __builtin_amdgcn_wmma_f32_16x16x4_f32(false, a0, false, b0,
                                                  (short)0, q, false, false);
        q = __builtin_amdgcn_wmma_f32_16x16x4_f32(false, a1, false, b1,
                                                  (short)0, q, false, false);

        // C/D layout: lane 0-15 -> M=r, lane 16-31 -> M=8+r; N = lane%16.
        const int gbase = g0 + half * 8;
        const float* rbase = realv + (size_t)gbase * NPIX + p;
#pragma unroll
        for (int r = 0; r < 8; ++r) {
            float rv   = rbase[(size_t)r * NPIX];   // coalesced 64B segments
            float s    = Sc[gbase + r];             // LDS broadcast
            float pred = s * __expf(-0.5f * q[r]);
            float e    = pred - rv;
            acc        = fmaf(e, e, acc);
        }
    }

    // ---- Reductions: wave32 shuffle -> cross-wave LDS -> block partial ----
#pragma unroll
    for (int off = 16; off > 0; off >>= 1)
        acc += __shfl_xor(acc, off, 32);
    if (lane == 0) red[wave] = acc;
    __syncthreads();
    if (t == 0) {
        float s = 0.0f;
#pragma unroll
        for (int i = 0; i < 8; ++i) s += red[i];
        ws[blockIdx.x] = s;
    }
}

// Deterministic final reduction of 512 block partials (no float atomics).
__global__ __launch_bounds__(256) void PlaceFields_reduce_kernel(
    const float* __restrict__ ws, float* __restrict__ out)
{
    __shared__ float s[256];
    const int t = threadIdx.x;
    s[t] = ws[t] + ws[t + 256];
    __syncthreads();
    for (int o = 128; o > 0; o >>= 1) {
        if (t < o) s[t] += s[t + o];
        __syncthreads();
    }
    if (t == 0) out[0] = s[0];
}

extern "C" void kernel_launch(void* const* d_in, const int* in_sizes, int n_in,
                              void* d_out, int out_size, void* d_ws, size_t ws_size,
                              hipStream_t stream) {
    const float* coords       = (const float*)d_in[0];
    const float* realv        = (const float*)d_in[1];
    const float* means        = (const float*)d_in[2];
    const float* cov_inv_diag = (const float*)d_in[3];
    const float* cov_inv_off  = (const float*)d_in[4];
    const float* scales       = (const float*)d_in[5];
    float* out = (float*)d_out;
    float* ws  = (float*)d_ws;   // 512 floats of block partials

    // 65536 pixels / (8 waves * 16 pixels) = 512 blocks
    PlaceFields_78932908966357_kernel<<<512, 256, 0, stream>>>(
        coords, realv, means, cov_inv_diag, cov_inv_off, scales, ws);
    PlaceFields_reduce_kernel<<<1, 256, 0, stream>>>(ws, out);
}